// EdgePredictor_48928267436424
// MI455X (gfx1250) — compile-verified
//
#include <hip/hip_runtime.h>

#define HIDDEN 128

typedef __attribute__((ext_vector_type(2))) float v2f;
typedef __attribute__((ext_vector_type(8))) float v8f;

// ---------------------------------------------------------------------------
// Kernel A:  P[n, 0:128]  = h[n,:] @ W1[  0:128, :]   (src projection)
//            P[n, 128:256]= h[n,:] @ W1[128:256, :]   (dst projection)
// 8 waves per workgroup; each wave computes one 16x16 tile with
// V_WMMA_F32_16X16X4_F32 (32 WMMAs over K=128). The 16-row A tile is staged
// once per WG into LDS with a 132-float row pitch (stride mod 64 banks = 4,
// so the per-lane 2-float reads cover all 64 banks conflict-free).
// ---------------------------------------------------------------------------
__global__ void __launch_bounds__(256)
gnn_gemm_kernel(const float* __restrict__ h,
                const float* __restrict__ W1,
                float* __restrict__ P,
                int n_nodes)
{
    __shared__ float As[16 * 132];

    const int wave = threadIdx.x >> 5;
    const int lane = threadIdx.x & 31;
    const int hl   = lane >> 4;       // half-wave select (K offset +0 / +2)
    const int l15  = lane & 15;

    const int tile = blockIdx.x * 8 + wave;  // 16 N-tiles per M-tile row;
    const int m0   = (tile >> 4) << 4;       // blocks of 8 tiles share one m0
    const int n0   = (tile & 15) << 4;

    // Cooperative stage of A rows m0..m0+15 (shared by all 8 waves).
    for (int i = threadIdx.x; i < 16 * HIDDEN; i += 256) {
        const int r  = i >> 7;       // row within tile
        const int c  = i & (HIDDEN - 1);
        const int gr = m0 + r;
        As[r * 132 + c] = (gr < n_nodes) ? h[(size_t)gr * HIDDEN + c] : 0.0f;
    }
    __syncthreads();

    // B operand: column (n0 & 127) + l15 of W1s (n0 < 128) or W1d (n0 >= 128)
    const int bbase = (n0 >= HIDDEN) ? HIDDEN : 0;
    const float* __restrict__ Bp =
        W1 + (size_t)bbase * HIDDEN + (n0 & (HIDDEN - 1)) + l15;

    v8f acc = {};

    #pragma unroll
    for (int kk = 0; kk < HIDDEN / 4; ++kk) {
        const int k0 = kk * 4 + hl * 2;     // lanes 0-15: K+0, lanes 16-31: K+2
        v2f a, b;
        a.x = As[l15 * 132 + k0];
        a.y = As[l15 * 132 + k0 + 1];
        b.x = Bp[(size_t)(k0 + 0) * HIDDEN];
        b.y = Bp[(size_t)(k0 + 1) * HIDDEN];
        // D = A(16x4,f32) * B(4x16,f32) + C(16x16,f32)
        acc = __builtin_amdgcn_wmma_f32_16x16x4_f32(
            /*neg_a=*/false, a, /*neg_b=*/false, b,
            /*c_mod=*/(short)0, acc, /*reuse_a=*/false, /*reuse_b=*/false);
    }

    // Store D: VGPR i -> row m0 + hl*8 + i, column n0 + l15.
    const int colP = n0 + l15;
    #pragma unroll
    for (int i = 0; i < 8; ++i) {
        const int r = m0 + hl * 8 + i;
        if (r < n_nodes)
            P[(size_t)r * (2 * HIDDEN) + colP] = acc[i];
    }
}

// ---------------------------------------------------------------------------
// Kernel B: one wave32 per edge.
//   x = P[src, 0:128] + P[dst, 128:256] + b1 ; relu ; score = x . W2 + b2
// Each lane handles 4 features (float4), butterfly-reduced across the wave.
// All gathered data (P = 102.4 MB) is L2-resident on MI455X (192 MB L2).
// ---------------------------------------------------------------------------
__global__ void __launch_bounds__(256)
gnn_edge_kernel(const float* __restrict__ P,
                const int*   __restrict__ edge,   // [2, n_edges] flat
                const float* __restrict__ b1,
                const float* __restrict__ W2,
                const float* __restrict__ b2,
                float*       __restrict__ out,
                int n_edges)
{
    const int e    = blockIdx.x * 8 + (threadIdx.x >> 5);
    const int lane = threadIdx.x & 31;
    if (e >= n_edges) return;

    const int src = edge[e];
    const int dst = edge[n_edges + e];

    const float4 ps = ((const float4*)(P + (size_t)src * (2 * HIDDEN)))[lane];
    const float4 pd = ((const float4*)(P + (size_t)dst * (2 * HIDDEN) + HIDDEN))[lane];
    const float4 bb = ((const float4*)b1)[lane];
    const float4 w  = ((const float4*)W2)[lane];

    float s = fmaxf(ps.x + pd.x + bb.x, 0.0f) * w.x
            + fmaxf(ps.y + pd.y + bb.y, 0.0f) * w.y
            + fmaxf(ps.z + pd.z + bb.z, 0.0f) * w.z
            + fmaxf(ps.w + pd.w + bb.w, 0.0f) * w.w;

    #pragma unroll
    for (int off = 16; off > 0; off >>= 1)
        s += __shfl_xor(s, off, 32);

    if (lane == 0) out[e] = s + b2[0];
}

// ---------------------------------------------------------------------------
extern "C" void kernel_launch(void* const* d_in, const int* in_sizes, int n_in,
                              void* d_out, int out_size, void* d_ws, size_t ws_size,
                              hipStream_t stream)
{
    const float* h  = (const float*)d_in[0];
    const int*   ei = (const int*)  d_in[1];
    const float* W1 = (const float*)d_in[2];
    const float* b1 = (const float*)d_in[3];
    const float* W2 = (const float*)d_in[4];
    const float* b2 = (const float*)d_in[5];
    float* out = (float*)d_out;
    float* P   = (float*)d_ws;            // [n_nodes][256] = 102.4 MB scratch

    const int n_nodes = in_sizes[0] / HIDDEN;   // 100000
    const int n_edges = in_sizes[1] / 2;        // 640000

    // Phase 1: node-level projections (WMMA GEMM). 2 blocks per 16-row M-tile
    // (16 N-tiles / 8 waves), so every block's 8 waves share one A tile.
    const int m_tiles = (n_nodes + 15) / 16;
    gnn_gemm_kernel<<<m_tiles * 2, 256, 0, stream>>>(h, W1, P, n_nodes);

    // Phase 2: per-edge combine + score (8 edges per 256-thread block).
    gnn_edge_kernel<<<(n_edges + 7) / 8, 256, 0, stream>>>(
        P, ei, b1, W2, b2, out, n_edges);
}